// GraphSAGE_90563680403608
// MI455X (gfx1250) — compile-verified
//
#include <hip/hip_runtime.h>
#include <hip/hip_bf16.h>
#include <math.h>

#define NNODES 50000
#define NEDGES 800000
#define DIN    128
#define HID    32

typedef __attribute__((ext_vector_type(2))) float v2f;
typedef __attribute__((ext_vector_type(8))) float v8f;

// ---------------------------------------------------------------------------
// Kernel 1: Y = X @ W^T for both W1_l and W1_r, via V_WMMA_F32_16X16X4_F32.
// One wave (32 threads) computes one 16x16 output tile.
// grid = (3125, 4): blockIdx.y bit1 selects weight/output, bit0 selects n-tile.
// ---------------------------------------------------------------------------
__global__ void __launch_bounds__(32)
sage_gemm_f32wmma(const float* __restrict__ x,
                  const float* __restrict__ W1l,
                  const float* __restrict__ W1r,
                  float* __restrict__ y_l,
                  float* __restrict__ y_r) {
    const int lane = threadIdx.x;                 // 0..31, wave32
    const int m0   = blockIdx.x << 4;             // node row tile
    const int n0   = (blockIdx.y & 1) << 4;       // output col tile (0 or 16)
    const bool right = (blockIdx.y & 2) != 0;
    const float* __restrict__ W = right ? W1r : W1l;       // [32,128] row-major
    float* __restrict__ yout    = right ? y_r : y_l;       // [N,32]

    // A 16x4 f32 layout: lanes 0-15 -> M=lane, VGPR0/1 = K 0/1; lanes 16-31 -> K 2/3
    const int mrow = lane & 15;
    const int koff = (lane >> 4) << 1;
    const float* __restrict__ arow = x + (size_t)(m0 + mrow) * DIN + koff;
    // B 4x16 f32 layout (K x N): lane&15 = N, lane>>4 selects K pair
    const float* __restrict__ brow = W + (size_t)(n0 + mrow) * DIN + koff;

    v8f c = {};
#pragma unroll
    for (int k0 = 0; k0 < DIN; k0 += 4) {
        v2f a, b;
        a.x = arow[k0];
        a.y = arow[k0 + 1];
        b.x = brow[k0];
        b.y = brow[k0 + 1];
        c = __builtin_amdgcn_wmma_f32_16x16x4_f32(
                /*neg_a=*/false, a, /*neg_b=*/false, b,
                /*c_mod=*/(short)0, c, /*reuse_a=*/false, /*reuse_b=*/false);
    }

    // C/D 16x16 f32 layout: lane L, VGPR v -> M = v + 8*(L>>4), N = L&15
    const int n = n0 + (lane & 15);
    const int mbase = m0 + ((lane >> 4) << 3);
#pragma unroll
    for (int v = 0; v < 8; ++v) {
        yout[(size_t)(mbase + v) * HID + n] = c[v];
    }
}

// ---------------------------------------------------------------------------
// Kernel 2: scatter-add projected neighbor features. 32 lanes per edge
// (lane = feature) -> coalesced atomics; lane 0 accumulates degree.
// ---------------------------------------------------------------------------
__global__ void __launch_bounds__(256)
sage_edge_agg32(const int* __restrict__ src,
                const int* __restrict__ dst,
                const float* __restrict__ y,
                float* __restrict__ agg,
                float* __restrict__ deg) {
    int t = blockIdx.x * blockDim.x + threadIdx.x;
    if (t >= NEDGES * HID) return;
    const int e = t >> 5;
    const int f = t & 31;
    const int s = src[e];
    const int d = dst[e];
    unsafeAtomicAdd(&agg[(size_t)d * HID + f], y[(size_t)s * HID + f]);
    if (f == 0) unsafeAtomicAdd(&deg[d], 1.0f);
}

// ---------------------------------------------------------------------------
// Kernel 3: h = relu(agg/max(deg,1) + b1 + y_r); fused layer-2 GEMV via
// wave32 shuffle reduction: z_l = h . W2_l, z_r = h . W2_r per node.
// 32 consecutive threads (one wave-aligned group) = one node.
// ---------------------------------------------------------------------------
__global__ void __launch_bounds__(256)
sage_act_fuse(const float* __restrict__ agg,
              const float* __restrict__ deg,
              const float* __restrict__ y_r,
              const float* __restrict__ b1,
              const float* __restrict__ W2l,
              const float* __restrict__ W2r,
              float* __restrict__ h,
              float* __restrict__ z_l,
              float* __restrict__ z_r) {
    int t = blockIdx.x * blockDim.x + threadIdx.x;
    if (t >= NNODES * HID) return;
    const int n = t >> 5;
    const int f = t & 31;
    const float dg = fmaxf(deg[n], 1.0f);
    float hv = agg[t] / dg + b1[f] + y_r[t];
    hv = fmaxf(hv, 0.0f);
    h[t] = hv;
    float pl = hv * W2l[f];
    float pr = hv * W2r[f];
#pragma unroll
    for (int off = 16; off > 0; off >>= 1) {
        pl += __shfl_xor(pl, off, 32);
        pr += __shfl_xor(pr, off, 32);
    }
    if (f == 0) {
        z_l[n] = pl;
        z_r[n] = pr;
    }
}

// ---------------------------------------------------------------------------
// Kernel 4: scalar layer-2 edge aggregation.
// ---------------------------------------------------------------------------
__global__ void __launch_bounds__(256)
sage_edge_agg1(const int* __restrict__ src,
               const int* __restrict__ dst,
               const float* __restrict__ z_l,
               float* __restrict__ agg2) {
    int e = blockIdx.x * blockDim.x + threadIdx.x;
    if (e >= NEDGES) return;
    unsafeAtomicAdd(&agg2[dst[e]], z_l[src[e]]);
}

// ---------------------------------------------------------------------------
// Kernel 5: out = sigmoid(agg2/max(deg,1) + b2 + z_r)
// ---------------------------------------------------------------------------
__global__ void __launch_bounds__(256)
sage_finalize(const float* __restrict__ agg2,
              const float* __restrict__ deg,
              const float* __restrict__ z_r,
              const float* __restrict__ b2,
              float* __restrict__ out) {
    int n = blockIdx.x * blockDim.x + threadIdx.x;
    if (n >= NNODES) return;
    const float dg = fmaxf(deg[n], 1.0f);
    const float v = agg2[n] / dg + b2[0] + z_r[n];
    out[n] = 1.0f / (1.0f + __expf(-v));
}

extern "C" void kernel_launch(void* const* d_in, const int* in_sizes, int n_in,
                              void* d_out, int out_size, void* d_ws, size_t ws_size,
                              hipStream_t stream) {
    // Inputs per setup_inputs() order
    const float* x   = (const float*)d_in[0];          // [N,128]
    const int*   ei  = (const int*)d_in[1];            // [2,E]
    const float* W1l = (const float*)d_in[2];          // [32,128]
    const float* b1  = (const float*)d_in[3];          // [32]
    const float* W1r = (const float*)d_in[4];          // [32,128]
    const float* W2l = (const float*)d_in[5];          // [1,32]
    const float* b2  = (const float*)d_in[6];          // [1]
    const float* W2r = (const float*)d_in[7];          // [1,32]
    float* out = (float*)d_out;                        // [N]

    const int* src = ei;
    const int* dst = ei + NEDGES;

    // Workspace layout (floats). agg/deg/agg2 kept contiguous for one memset.
    float* ws   = (float*)d_ws;
    float* y_l  = ws;                                  // N*32
    float* y_r  = y_l + (size_t)NNODES * HID;          // N*32
    float* agg  = y_r + (size_t)NNODES * HID;          // N*32  (zeroed)
    float* deg  = agg + (size_t)NNODES * HID;          // N     (zeroed)
    float* agg2 = deg + NNODES;                        // N     (zeroed)
    float* h    = agg2 + NNODES;                       // N*32
    float* z_l  = h + (size_t)NNODES * HID;            // N
    float* z_r  = z_l + NNODES;                        // N

    // Zero the accumulators (agg | deg | agg2 are contiguous)
    hipMemsetAsync(agg, 0, ((size_t)NNODES * HID + 2 * (size_t)NNODES) * sizeof(float),
                   stream);

    // 1) y_l = x@W1_l.T, y_r = x@W1_r.T  (WMMA f32 16x16x4)
    dim3 ggrid(NNODES / 16, 4);
    sage_gemm_f32wmma<<<ggrid, 32, 0, stream>>>(x, W1l, W1r, y_l, y_r);

    // 2) agg[dst] += y_l[src]; deg[dst] += 1
    int t2 = NEDGES * HID;
    sage_edge_agg32<<<(t2 + 255) / 256, 256, 0, stream>>>(src, dst, y_l, agg, deg);

    // 3) h = relu(agg/deg + b1 + y_r); z_l = h.W2l; z_r = h.W2r
    int t3 = NNODES * HID;
    sage_act_fuse<<<(t3 + 255) / 256, 256, 0, stream>>>(agg, deg, y_r, b1, W2l, W2r,
                                                        h, z_l, z_r);

    // 4) agg2[dst] += z_l[src]
    sage_edge_agg1<<<(NEDGES + 255) / 256, 256, 0, stream>>>(src, dst, z_l, agg2);

    // 5) out = sigmoid(agg2/deg + b2 + z_r)
    sage_finalize<<<(NNODES + 255) / 256, 256, 0, stream>>>(agg2, deg, z_r, b2, out);
}